// GATNet_54030688584327
// MI455X (gfx1250) — compile-verified
//
#include <hip/hip_runtime.h>
#include <hip/hip_bf16.h>
#include <math.h>

typedef float v2f __attribute__((ext_vector_type(2)));
typedef float v8f __attribute__((ext_vector_type(8)));

// ---- order-preserving float <-> uint encoding for atomic float max ----
__device__ __forceinline__ unsigned f2ord(float f) {
    int i = __float_as_int(f);
    return (i >= 0) ? ((unsigned)i | 0x80000000u) : ~((unsigned)i);
}
__device__ __forceinline__ float ord2f(unsigned u) {
    int i = (u & 0x80000000u) ? (int)(u & 0x7fffffffu) : (int)(~u);
    return __int_as_float(i);
}

__global__ void zero_kernel(float* __restrict__ p, long long n) {
    long long i = blockIdx.x * (long long)blockDim.x + threadIdx.x;
    long long stride = (long long)gridDim.x * blockDim.x;
    for (; i < n; i += stride) p[i] = 0.f;
}

// ---------------------------------------------------------------------------
// FP32 GEMM via V_WMMA_F32_16X16X4_F32.
// One wave computes a 16-row strip by NT 16-col tiles; A fragment reused
// across all NT column tiles. FULL=true path has no per-lane predication in
// the k-loop and no guarded stores (taken whenever the 16-row tile is full,
// which is always the case here since M % 16 == 0).
// ---------------------------------------------------------------------------
template <int NT, bool FULL>
__device__ __forceinline__ void gemm_wmma_body(const float* __restrict__ A,
                                               const float* __restrict__ B,
                                               float* __restrict__ C,
                                               const float* __restrict__ bias,
                                               int M, int K, int ldb, int ldc,
                                               int rowTile, int colBase, int lane) {
    v8f acc[NT];
#pragma unroll
    for (int j = 0; j < NT; ++j)
#pragma unroll
        for (int r = 0; r < 8; ++r) acc[j][r] = 0.f;

    // A frag: lanes 0-15 -> K = k0+0..1 ; lanes 16-31 -> K = k0+2..3
    int row     = rowTile * 16 + (lane & 15);
    int koff    = (lane >> 4) << 1;
    int colLane = lane & 15;
    const float* Arow = A + (size_t)row * K + koff;

    for (int k0 = 0; k0 < K; k0 += 4) {
        v2f a;
        if (FULL) {
            a = *(const v2f*)(Arow + k0);
        } else {
            a = (v2f){0.f, 0.f};
            if (row < M) a = *(const v2f*)(Arow + k0);
        }
#pragma unroll
        for (int j = 0; j < NT; ++j) {
            int col = colBase + j * 16 + colLane;
            v2f b;
            b[0] = B[(size_t)(k0 + koff)     * ldb + col];
            b[1] = B[(size_t)(k0 + koff + 1) * ldb + col];
            acc[j] = __builtin_amdgcn_wmma_f32_16x16x4_f32(
                false, a, false, b, (short)0, acc[j], false, false);
        }
    }

    // C/D layout: VGPR r -> M = r (lanes 0-15) or r+8 (lanes 16-31), N = lane&15
    int mBase = rowTile * 16 + ((lane >> 4) << 3);
#pragma unroll
    for (int j = 0; j < NT; ++j) {
        int col = colBase + j * 16 + colLane;
        float bj = bias ? bias[col] : 0.f;
#pragma unroll
        for (int r = 0; r < 8; ++r) {
            int m = mBase + r;
            if (FULL) {
                C[(size_t)m * ldc + col] = acc[j][r] + bj;
            } else if (m < M) {
                C[(size_t)m * ldc + col] = acc[j][r] + bj;
            }
        }
    }
}

template <int NT>
__global__ void gemm_wmma_kernel(const float* __restrict__ A,
                                 const float* __restrict__ B,
                                 float* __restrict__ C,
                                 const float* __restrict__ bias,
                                 int M, int K, int ldb, int ldc,
                                 int colGroups) {
    int wid  = (int)((blockIdx.x * (long long)blockDim.x + threadIdx.x) >> 5);
    int lane = threadIdx.x & 31;
    int rowTile = wid / colGroups;
    int group   = wid - rowTile * colGroups;
    int rowTiles = (M + 15) >> 4;
    if (rowTile >= rowTiles) return;           // uniform per wave
    int colBase = group * (NT * 16);

    if (rowTile * 16 + 16 <= M) {              // uniform branch, always taken here
        gemm_wmma_body<NT, true >(A, B, C, bias, M, K, ldb, ldc, rowTile, colBase, lane);
    } else {
        gemm_wmma_body<NT, false>(A, B, C, bias, M, K, ldb, ldc, rowTile, colBase, lane);
    }
}

// one wave per (node, head): alpha_s/alpha_d = dot(h[n,h,:], a_src/a_dst[h,:])
__global__ void alpha_kernel(const float* __restrict__ h,
                             const float* __restrict__ a_src,
                             const float* __restrict__ a_dst,
                             float* __restrict__ as, float* __restrict__ ad,
                             int N, int C) {
    int wid  = (int)((blockIdx.x * (long long)blockDim.x + threadIdx.x) >> 5);
    int lane = threadIdx.x & 31;
    if (wid >= N * 4) return;
    int n = wid >> 2, hh = wid & 3;
    const float* hp  = h + (size_t)n * 4 * C + (size_t)hh * C;
    const float* asv = a_src + hh * C;
    const float* adv = a_dst + hh * C;
    float s = 0.f, d = 0.f;
    for (int c = lane; c < C; c += 32) {
        float v = hp[c];
        s += v * asv[c];
        d += v * adv[c];
    }
#pragma unroll
    for (int off = 16; off; off >>= 1) {
        s += __shfl_xor(s, off, 32);
        d += __shfl_xor(d, off, 32);
    }
    if (lane == 0) { as[n * 4 + hh] = s; ad[n * 4 + hh] = d; }
}

__device__ __forceinline__ void edge_sd(int e, int E, const long long* src,
                                        const long long* dst, int& s, int& d) {
    if (e < E) { s = (int)src[e]; d = (int)dst[e]; }
    else       { s = d = e - E; }               // self loops appended
}

__global__ void edge_max_kernel(const float* __restrict__ as,
                                const float* __restrict__ ad,
                                const long long* __restrict__ src,
                                const long long* __restrict__ dst,
                                unsigned* __restrict__ emax, int E, int Etot) {
    int idx = blockIdx.x * blockDim.x + threadIdx.x;
    int e = idx >> 2, hh = idx & 3;
    if (e >= Etot) return;
    int s, d; edge_sd(e, E, src, dst, s, d);
    float v = as[s * 4 + hh] + ad[d * 4 + hh];
    v = (v > 0.f) ? v : 0.2f * v;               // leaky_relu(0.2)
    atomicMax(&emax[d * 4 + hh], f2ord(v));
}

__global__ void edge_exp_kernel(const float* __restrict__ as,
                                const float* __restrict__ ad,
                                const long long* __restrict__ src,
                                const long long* __restrict__ dst,
                                const unsigned* __restrict__ emax,
                                float* __restrict__ denom,
                                float* __restrict__ eexp, int E, int Etot) {
    int idx = blockIdx.x * blockDim.x + threadIdx.x;
    int e = idx >> 2, hh = idx & 3;
    if (e >= Etot) return;
    int s, d; edge_sd(e, E, src, dst, s, d);
    float v = as[s * 4 + hh] + ad[d * 4 + hh];
    v = (v > 0.f) ? v : 0.2f * v;
    float w = expf(v - ord2f(emax[d * 4 + hh]));
    eexp[(size_t)e * 4 + hh] = w;
    atomicAdd(&denom[d * 4 + hh], w);
}

// scatter: one thread per (edge, channel); HCshift=log2(H*C), Cshift=log2(C)
__global__ void agg_kernel(const float* __restrict__ hfeat,
                           const float* __restrict__ eexp,
                           const float* __restrict__ denom,
                           const long long* __restrict__ src,
                           const long long* __restrict__ dst,
                           float* __restrict__ acc,
                           int E, int Etot, int HCshift, int Cshift) {
    long long gid = blockIdx.x * (long long)blockDim.x + threadIdx.x;
    int e = (int)(gid >> HCshift);
    if (e >= Etot) return;
    int t  = (int)(gid & ((1 << HCshift) - 1));
    int hh = t >> Cshift;
    int s, d; edge_sd(e, E, src, dst, s, d);
    float alpha = eexp[(size_t)e * 4 + hh] / (denom[d * 4 + hh] + 1e-16f);
    atomicAdd(&acc[((size_t)d << HCshift) + t],
              hfeat[((size_t)s << HCshift) + t] * alpha);
}

__global__ void bias_elu_kernel(float* __restrict__ acc,
                                const float* __restrict__ b, long long n) {
    long long i = blockIdx.x * (long long)blockDim.x + threadIdx.x;
    if (i >= n) return;
    float v = acc[i] + b[(int)(i & 127)];
    acc[i] = (v > 0.f) ? v : (expf(v) - 1.f);   // ELU(alpha=1)
}

__global__ void mean_bias_kernel(const float* __restrict__ acc2,
                                 const float* __restrict__ b2,
                                 float* __restrict__ emb, long long n) {
    long long i = blockIdx.x * (long long)blockDim.x + threadIdx.x;
    if (i >= n) return;
    int nn = (int)(i >> 7), c = (int)(i & 127);
    const float* p = acc2 + (size_t)nn * 512 + c;
    float s = p[0] + p[128] + p[256] + p[384];
    emb[i] = s * 0.25f + b2[c];
}

extern "C" void kernel_launch(void* const* d_in, const int* in_sizes, int n_in,
                              void* d_out, int out_size, void* d_ws, size_t ws_size,
                              hipStream_t stream) {
    const float*     x      = (const float*)d_in[0];
    const long long* ei     = (const long long*)d_in[1];   // int64 [2,E]
    const float*     W1     = (const float*)d_in[2];
    const float*     a_src1 = (const float*)d_in[3];
    const float*     a_dst1 = (const float*)d_in[4];
    const float*     b1     = (const float*)d_in[5];
    const float*     W2     = (const float*)d_in[6];
    const float*     a_src2 = (const float*)d_in[7];
    const float*     a_dst2 = (const float*)d_in[8];
    const float*     b2     = (const float*)d_in[9];
    const float*     W_lin  = (const float*)d_in[10];
    const float*     b_lin  = (const float*)d_in[11];

    const int N    = in_sizes[0] / 512;
    const int E    = in_sizes[1] / 2;
    const int Etot = E + N;                    // + self loops

    // workspace layout (floats): ~530 MB total
    float* ws    = (float*)d_ws;
    float* h1    = ws;                          // N*128
    float* acc1  = h1   + (size_t)N * 128;      // N*128 (becomes ELU output)
    float* h3    = acc1 + (size_t)N * 128;      // N*512
    float* acc2  = h3   + (size_t)N * 512;      // N*512
    float* as    = acc2 + (size_t)N * 512;      // N*4
    float* ad    = as   + (size_t)N * 4;        // N*4
    float* denom = ad   + (size_t)N * 4;        // N*4
    unsigned* emax = (unsigned*)(denom + (size_t)N * 4);   // N*4
    float* eexp  = (float*)(emax + (size_t)N * 4);         // Etot*4

    const long long* srcI = ei;
    const long long* dstI = ei + E;

    float* outp = (float*)d_out;                // N*64 logits
    float* emb  = outp + (size_t)N * 64;        // N*128 embeddings

    auto blocks = [](long long n) { return (unsigned)((n + 255) / 256); };
    int rowTiles = (N + 15) / 16;

    // ---------------- layer 1 ----------------
    gemm_wmma_kernel<8><<<blocks((long long)rowTiles * 32), 256, 0, stream>>>(
        x, W1, h1, nullptr, N, 512, 128, 128, 1);

    alpha_kernel<<<blocks((long long)N * 4 * 32), 256, 0, stream>>>(
        h1, a_src1, a_dst1, as, ad, N, 32);

    zero_kernel<<<blocks((long long)N * 4), 256, 0, stream>>>((float*)emax, (long long)N * 4);
    zero_kernel<<<blocks((long long)N * 4), 256, 0, stream>>>(denom, (long long)N * 4);
    zero_kernel<<<blocks((long long)N * 128), 256, 0, stream>>>(acc1, (long long)N * 128);

    edge_max_kernel<<<blocks((long long)Etot * 4), 256, 0, stream>>>(
        as, ad, srcI, dstI, emax, E, Etot);
    edge_exp_kernel<<<blocks((long long)Etot * 4), 256, 0, stream>>>(
        as, ad, srcI, dstI, emax, denom, eexp, E, Etot);
    agg_kernel<<<blocks((long long)Etot << 7), 256, 0, stream>>>(
        h1, eexp, denom, srcI, dstI, acc1, E, Etot, 7, 5);

    bias_elu_kernel<<<blocks((long long)N * 128), 256, 0, stream>>>(
        acc1, b1, (long long)N * 128);

    // ---------------- layer 2 ----------------
    gemm_wmma_kernel<8><<<blocks((long long)rowTiles * 4 * 32), 256, 0, stream>>>(
        acc1, W2, h3, nullptr, N, 128, 512, 512, 4);

    alpha_kernel<<<blocks((long long)N * 4 * 32), 256, 0, stream>>>(
        h3, a_src2, a_dst2, as, ad, N, 128);

    zero_kernel<<<blocks((long long)N * 4), 256, 0, stream>>>((float*)emax, (long long)N * 4);
    zero_kernel<<<blocks((long long)N * 4), 256, 0, stream>>>(denom, (long long)N * 4);
    zero_kernel<<<blocks((long long)N * 512), 256, 0, stream>>>(acc2, (long long)N * 512);

    edge_max_kernel<<<blocks((long long)Etot * 4), 256, 0, stream>>>(
        as, ad, srcI, dstI, emax, E, Etot);
    edge_exp_kernel<<<blocks((long long)Etot * 4), 256, 0, stream>>>(
        as, ad, srcI, dstI, emax, denom, eexp, E, Etot);
    agg_kernel<<<blocks((long long)Etot << 9), 256, 0, stream>>>(
        h3, eexp, denom, srcI, dstI, acc2, E, Etot, 9, 7);

    mean_bias_kernel<<<blocks((long long)N * 128), 256, 0, stream>>>(
        acc2, b2, emb, (long long)N * 128);

    // ---------------- linear head ----------------
    gemm_wmma_kernel<4><<<blocks((long long)rowTiles * 32), 256, 0, stream>>>(
        emb, W_lin, outp, b_lin, N, 128, 64, 64, 1);
}